// IrregularTimeAttention_48988396978195
// MI455X (gfx1250) — compile-verified
//
#include <hip/hip_runtime.h>
#include <hip/hip_bf16.h>
#include <cstdint>

typedef __attribute__((ext_vector_type(2))) float v2f;
typedef __attribute__((ext_vector_type(8))) float v8f;

#define NB   4
#define NH   4
#define NLQ  512
#define NLK  512
#define NDK  64
#define NDIM 32
#define NEGV (-1e9f)

// One block per (b,h, 16-row q tile). 512 threads = 16 waves (wave32).
__global__ __launch_bounds__(512)
void attn_wmma_kernel(const float* __restrict__ qg,
                      const float* __restrict__ kg,
                      const float* __restrict__ vg,
                      const uint8_t* __restrict__ mask,
                      float* __restrict__ outg,
                      float* __restrict__ pattn)
{
    __shared__ float sS[16 * NLK];    // 32 KB: score tile, scaled
    __shared__ float sV[NLK * NDIM];  // 64 KB: V tile for this (b,h)

    const int t    = threadIdx.x;
    const int lane = t & 31;
    const int row  = lane & 15;   // M (A) / N (B,C,D) index within 16
    const int hi   = lane >> 4;   // lane-half select

    const int bh = blockIdx.x >> 5;   // 32 q-tiles per (b,h)
    const int qt = blockIdx.x & 31;
    const int q0 = qt * 16;
    const int b  = bh >> 2;           // H == 4

    const float* qp = qg + ((size_t)bh * NLQ + q0) * NDK;
    const float* kp = kg + (size_t)bh * NLK * NDK;
    const float* vp = vg + (size_t)bh * NLK * NDIM;

    // ---- stage V into LDS via async global->LDS DMA (no VGPR round-trip;
    //      overlaps with the WMMA phase; tracked by ASYNCcnt) ----
    {
        const uint32_t sVbase = (uint32_t)(uintptr_t)(void*)sV; // LDS byte offset
        for (int idx = t; idx < NLK * NDIM / 4; idx += 512) {
            const uint64_t ga = (uint64_t)(uintptr_t)(vp + idx * 4);
            const uint32_t la = sVbase + (uint32_t)idx * 16u;
            asm volatile("global_load_async_to_lds_b128 %0, %1, off"
                         :: "v"(la), "v"(ga) : "memory");
        }
    }

    // ---- S = (Q K^T) * 1/sqrt(64) via fp32 WMMA 16x16x4 ----
    // wave w handles k-tiles w and w+16 (32 tiles total)
    for (int kt = t >> 5; kt < NLK / 16; kt += 16) {
        const int k0 = kt * 16;
        v8f acc = {};
        #pragma unroll
        for (int d0 = 0; d0 < NDK; d0 += 4) {
            // A 16x4 f32: VGPR0 = K=0 (lanes 0-15) / K=2 (lanes 16-31), VGPR1 = K=1 / K=3
            const int d = d0 + 2 * hi;
            v2f a, bb;
            a.x  = qp[row * NDK + d];
            a.y  = qp[row * NDK + d + 1];
            // B 4x16: element (K=d, N=row) = key[k0+row][d]
            bb.x = kp[(k0 + row) * NDK + d];
            bb.y = kp[(k0 + row) * NDK + d + 1];
            acc = __builtin_amdgcn_wmma_f32_16x16x4_f32(
                false, a, false, bb, (short)0, acc, false, false);
        }
        // C/D layout: VGPR i -> M = i (lanes 0-15) / M = i+8 (lanes 16-31), N = row
        #pragma unroll
        for (int i = 0; i < 8; ++i)
            sS[(i + 8 * hi) * NLK + k0 + row] = acc[i] * 0.125f;
    }

    // V async copies must have landed in LDS before any wave reads them.
    asm volatile("s_wait_asynccnt 0x0" ::: "memory");
    __syncthreads();

    // ---- per-(q,c) masked online softmax over k, mask read exactly once ----
    const int qr = t >> 5;   // 0..15 : all lanes of a wave share qr -> LDS broadcast
    const int c  = t & 31;   // coalesced dimension for mask / p_attn / v
    const float* srow = &sS[qr * NLK];
    const uint8_t* mrow =
        mask + ((size_t)b * NLQ + (q0 + qr)) * NLK * NDIM + c;

    uint32_t mbits[NLK / 32];
    float mx = -3.4e38f, sum = 0.f;
    #pragma unroll 1
    for (int kk = 0; kk < NLK / 32; ++kk) {
        uint32_t bits = 0;
        for (int j = 0; j < 32; ++j) {
            const int k = kk * 32 + j;
            const uint32_t mb = (uint32_t)(mrow[(size_t)k * NDIM]) & 1u;
            bits |= mb << j;
            const float s = mb ? srow[k] : NEGV;
            if (s > mx) { sum *= __expf(mx - s); mx = s; }
            sum += __expf(s - mx);
        }
        mbits[kk] = bits;
    }
    const float inv = 1.0f / sum;

    // ---- write p_attn (dominant 537 MB stream, coalesced) + accumulate out ----
    float acco = 0.f;
    float* prow = pattn + ((size_t)bh * NLQ + (q0 + qr)) * NLK * NDIM + c;
    #pragma unroll 1
    for (int kk = 0; kk < NLK / 32; ++kk) {
        const uint32_t bits = mbits[kk];
        for (int j = 0; j < 32; ++j) {
            const int k = kk * 32 + j;
            const float s = ((bits >> j) & 1u) ? srow[k] : NEGV;
            const float p = __expf(s - mx) * inv;
            prow[(size_t)k * NDIM] = p;
            acco += p * sV[k * NDIM + c];
        }
    }
    outg[((size_t)bh * NLQ + (q0 + qr)) * NDIM + c] = acco;
}

extern "C" void kernel_launch(void* const* d_in, const int* in_sizes, int n_in,
                              void* d_out, int out_size, void* d_ws, size_t ws_size,
                              hipStream_t stream) {
    (void)in_sizes; (void)n_in; (void)d_ws; (void)ws_size; (void)out_size;
    const float*   q  = (const float*)d_in[0];
    const float*   k  = (const float*)d_in[1];
    const float*   v  = (const float*)d_in[2];
    const uint8_t* m  = (const uint8_t*)d_in[3];   // jnp.bool_ -> 1 byte/elt

    float* out   = (float*)d_out;                              // [B,H,LQ,DIM]
    float* pattn = out + (size_t)NB * NH * NLQ * NDIM;         // [B,H,LQ,LK,DIM]

    const int grid = NB * NH * (NLQ / 16);  // 512 blocks
    attn_wmma_kernel<<<grid, 512, 0, stream>>>(q, k, v, m, out, pattn);
}